// L2_Cognitive_Schema_Vault_89644557402496
// MI455X (gfx1250) — compile-verified
//
#include <hip/hip_runtime.h>
#include <math.h>

// ---------------- types ----------------
typedef __attribute__((ext_vector_type(2))) float v2f;
typedef __attribute__((ext_vector_type(8))) float v8f;
typedef __attribute__((ext_vector_type(4))) int v4i;

#define N_VAULT   2000000
#define DIMS      128
#define NB_SUM    1024
#define NB_SCORE  1024
#define TILE_R    64
#define NTILES    (N_VAULT / TILE_R)   // 31250
#define SC_STR    67                   // conflict-free stride for score LDS

// ---------------- CDNA5 async copy (global -> LDS), ASYNCcnt tracked ----------------
#if __has_builtin(__builtin_amdgcn_global_load_async_to_lds_b128)
#define ASYNC_CP16(g, l)                                                     \
  __builtin_amdgcn_global_load_async_to_lds_b128(                            \
      (__attribute__((address_space(1))) v4i*)(g),                           \
      (__attribute__((address_space(3))) v4i*)(l), 0, 0)
#else
__device__ __forceinline__ void __async_cp16_asm(const void* g, void* l) {
  unsigned lds =
      (unsigned)(unsigned long long)(__attribute__((address_space(3))) void*)l;
  asm volatile("global_load_async_to_lds_b128 %0, %1, off"
               :: "v"(lds),
                  "v"((__attribute__((address_space(1))) void*)g)
               : "memory");
}
#define ASYNC_CP16(g, l) __async_cp16_asm((g), (l))
#endif

#if __has_builtin(__builtin_amdgcn_s_wait_asynccnt)
#define WAIT_ASYNC(n) __builtin_amdgcn_s_wait_asynccnt(n)
#else
#define WAIT_ASYNC(n) asm volatile("s_wait_asynccnt %0" :: "i"(n) : "memory")
#endif

// ---------------- deterministic top-8 insertion (val desc, idx asc on ties) -------
__device__ __forceinline__ void topk_insert(float (&tv)[8], int (&ti)[8],
                                            float s, int idx) {
  bool notBetter = (s < tv[7]) || (s == tv[7] && idx >= ti[7]);
  if (notBetter) return;
  tv[7] = s; ti[7] = idx;
#pragma unroll
  for (int k = 7; k > 0; --k) {
    bool sw = (tv[k] > tv[k - 1]) || (tv[k] == tv[k - 1] && ti[k] < ti[k - 1]);
    if (sw) {
      float fv = tv[k]; tv[k] = tv[k - 1]; tv[k - 1] = fv;
      int iv = ti[k]; ti[k] = ti[k - 1]; ti[k - 1] = iv;
    }
  }
}

// ---------------- kernel 1: per-block partial column sums of vault ----------------
__global__ __launch_bounds__(256) void mean_kernel(const float* __restrict__ vault,
                                                   float* __restrict__ bsum) {
  __shared__ float red[256];
  const int tid = threadIdx.x;
  const int d = tid & 127, half = tid >> 7;
  float acc = 0.f;
  for (long long r = (long long)blockIdx.x * 2 + half; r < N_VAULT;
       r += 2LL * NB_SUM)
    acc += vault[r * DIMS + d];
  red[tid] = acc;
  __syncthreads();
  if (tid < 128) bsum[blockIdx.x * DIMS + tid] = red[tid] + red[tid + 128];
}

// ---------------- kernel 2: finalize a, build Qext (80x128) and aq -----------------
__global__ __launch_bounds__(128) void prep_kernel(const float* __restrict__ x,
                                                   const float* __restrict__ bsum,
                                                   float* __restrict__ qext,
                                                   float* __restrict__ aq) {
  __shared__ float red[128], aL[128], atL[128], qL[128], xL[128];
  const int tid = threadIdx.x;
  float s = 0.f;
  for (int i = 0; i < NB_SUM; ++i) s += bsum[i * DIMS + tid];
  float a = s * (1.0f / (float)N_VAULT);
  red[tid] = a * a;
  __syncthreads();
  for (int off = 64; off > 0; off >>= 1) {
    if (tid < off) red[tid] += red[tid + off];
    __syncthreads();
  }
  a /= sqrtf(red[0] + 1e-8f);
  aL[tid] = a;
  __syncthreads();
  float at = ((tid & 1) == 0) ? aL[tid + 1] : -aL[tid - 1];
  atL[tid] = at;
  qext[64 * DIMS + tid] = a;   // va_re row
  qext[65 * DIMS + tid] = at;  // va_im row
  for (int r = 66; r < 80; ++r) qext[r * DIMS + tid] = 0.f;
  __syncthreads();

  for (int b = 0; b < 32; ++b) {
    float xv = x[b * DIMS + tid];
    xL[tid] = xv;
    __syncthreads();
    int g = tid & ~3;
    float s4 = xL[g] * xL[g] + xL[g + 1] * xL[g + 1] + xL[g + 2] * xL[g + 2] +
               xL[g + 3] * xL[g + 3];
    float q = xv / sqrtf(s4 + 1e-8f);
    qL[tid] = q;
    __syncthreads();
    float qt = ((tid & 1) == 0) ? -qL[tid + 1] : qL[tid - 1];
    qext[b * DIMS + tid] = q;          // qv_re rows
    qext[(32 + b) * DIMS + tid] = qt;  // qv_im rows
    // aq_re[b] = q·a ; aq_im[b] = -(q·ã)
    red[tid] = q * aL[tid];
    __syncthreads();
    for (int off = 64; off > 0; off >>= 1) {
      if (tid < off) red[tid] += red[tid + off];
      __syncthreads();
    }
    if (tid == 0) aq[b] = red[0];
    __syncthreads();
    red[tid] = q * atL[tid];
    __syncthreads();
    for (int off = 64; off > 0; off >>= 1) {
      if (tid < off) red[tid] += red[tid + off];
      __syncthreads();
    }
    if (tid == 0) aq[32 + b] = -red[0];
    __syncthreads();
  }
}

// ---------------- kernel 3: WMMA scoring + per-block top-8 -------------------------
__global__ __launch_bounds__(128) void score_kernel(
    const float* __restrict__ vault, const float* __restrict__ qext_g,
    const float* __restrict__ aq_g, float* __restrict__ candV,
    int* __restrict__ candI) {
  __shared__ float tile[2][TILE_R * DIMS];  // 2 x 32KB, async double buffer
  __shared__ float qe[80 * DIMS];           // 40KB Qext
  __shared__ float aqs[64];
  __shared__ float sc[32 * SC_STR];         // per-tile score staging

  const int tid = threadIdx.x;
  for (int i = tid; i < 80 * DIMS; i += 128) qe[i] = qext_g[i];
  if (tid < 64) aqs[tid] = aq_g[tid];

  float tv[8];
  int ti[8];
#pragma unroll
  for (int k = 0; k < 8; ++k) { tv[k] = -3.402823e38f; ti[k] = 0x7fffffff; }

  const int wave = tid >> 5, lane = tid & 31;
  const int hi = lane >> 4, ln = lane & 15;
  const int jb = wave * 16;  // each of 4 waves owns one 16-row N tile
  const int col = 2 * hi;

  // prologue: async-fetch first tile
  {
    int t0 = blockIdx.x;
    if (t0 < NTILES) {
      const float* src = vault + (long long)t0 * TILE_R * DIMS;
#pragma unroll
      for (int i = 0; i < 16; ++i)
        ASYNC_CP16(src + (i * 128 + tid) * 4, &tile[0][(i * 128 + tid) * 4]);
    }
  }
  __syncthreads();

  int cur = 0;
  for (int t = blockIdx.x; t < NTILES; t += NB_SCORE) {
    int tn = t + NB_SCORE;
    if (tn < NTILES) {  // prefetch next tile into the other buffer
      const float* src = vault + (long long)tn * TILE_R * DIMS;
#pragma unroll
      for (int i = 0; i < 16; ++i)
        ASYNC_CP16(src + (i * 128 + tid) * 4,
                   &tile[cur ^ 1][(i * 128 + tid) * 4]);
      WAIT_ASYNC(16);  // current tile's 16 loads done; next 16 in flight
    } else {
      WAIT_ASYNC(0);
    }
    __syncthreads();

    const float* tb = &tile[cur][0];
    v8f c0 = {}, c1 = {}, c2 = {}, c3 = {}, c4 = {};
    const float* brow = tb + (jb + ln) * DIMS + col;       // B fragment base
    const float* a0p = qe + (0 * 16 + ln) * DIMS + col;    // qv_re b 0..15
    const float* a1p = qe + (1 * 16 + ln) * DIMS + col;    // qv_re b 16..31
    const float* a2p = qe + (2 * 16 + ln) * DIMS + col;    // qv_im b 0..15
    const float* a3p = qe + (3 * 16 + ln) * DIMS + col;    // qv_im b 16..31
    const float* a4p = qe + (4 * 16 + ln) * DIMS + col;    // va rows
#pragma unroll 8
    for (int kc = 0; kc < 32; ++kc) {
      v2f bf = *(const v2f*)(brow + 4 * kc);
      v2f a0 = *(const v2f*)(a0p + 4 * kc);
      c0 = __builtin_amdgcn_wmma_f32_16x16x4_f32(false, a0, false, bf,
                                                 (short)0, c0, false, false);
      v2f a1 = *(const v2f*)(a1p + 4 * kc);
      c1 = __builtin_amdgcn_wmma_f32_16x16x4_f32(false, a1, false, bf,
                                                 (short)0, c1, false, false);
      v2f a2 = *(const v2f*)(a2p + 4 * kc);
      c2 = __builtin_amdgcn_wmma_f32_16x16x4_f32(false, a2, false, bf,
                                                 (short)0, c2, false, false);
      v2f a3 = *(const v2f*)(a3p + 4 * kc);
      c3 = __builtin_amdgcn_wmma_f32_16x16x4_f32(false, a3, false, bf,
                                                 (short)0, c3, false, false);
      v2f a4 = *(const v2f*)(a4p + 4 * kc);
      c4 = __builtin_amdgcn_wmma_f32_16x16x4_f32(false, a4, false, bf,
                                                 (short)0, c4, false, false);
    }

    // va_re (M=0) / va_im (M=1) live in lanes 0-15 of c4; broadcast to hi half
    float vaRe = __shfl(c4[0], ln, 32);
    float vaIm = __shfl(c4[1], ln, 32);
    const int jj = jb + ln;
#pragma unroll
    for (int v = 0; v < 8; ++v) {
      int bA = v + 8 * hi, bB = bA + 16;
      float reA = c0[v], imA = c2[v];
      float trA = reA * vaRe - imA * vaIm;
      float tiA_ = reA * vaIm + imA * vaRe;
      float b3A = trA * aqs[bA] - tiA_ * aqs[32 + bA];
      sc[bA * SC_STR + jj] = 0.5f * (reA + b3A);
      float reB = c1[v], imB = c3[v];
      float trB = reB * vaRe - imB * vaIm;
      float tiB_ = reB * vaIm + imB * vaRe;
      float b3B = trB * aqs[bB] - tiB_ * aqs[32 + bB];
      sc[bB * SC_STR + jj] = 0.5f * (reB + b3B);
    }
    __syncthreads();

    if (wave == 0) {  // lane = b : scan this tile's 64 scores, keep top-8
      int base = t * TILE_R;
      for (int j2 = 0; j2 < TILE_R; ++j2)
        topk_insert(tv, ti, sc[lane * SC_STR + j2], base + j2);
    }
    __syncthreads();
    cur ^= 1;
  }

  if (wave == 0) {
#pragma unroll
    for (int k = 0; k < 8; ++k) {
      candV[blockIdx.x * 256 + lane * 8 + k] = tv[k];
      candI[blockIdx.x * 256 + lane * 8 + k] = ti[k];
    }
  }
}

// ---------------- kernel 4: global top-8 merge, softmax, gather --------------------
__global__ __launch_bounds__(128) void merge_kernel(
    const float* __restrict__ vault, const float* __restrict__ candV,
    const int* __restrict__ candI, float* __restrict__ out) {
  __shared__ float lv[128 * 8];
  __shared__ int li[128 * 8];
  __shared__ float w8[8];
  __shared__ int i8[8];
  const int b = blockIdx.x, tid = threadIdx.x;
  float tv[8];
  int ti[8];
#pragma unroll
  for (int k = 0; k < 8; ++k) { tv[k] = -3.402823e38f; ti[k] = 0x7fffffff; }
  for (int i = tid; i < NB_SCORE * 8; i += 128) {
    int blk = i >> 3, k = i & 7;
    topk_insert(tv, ti, candV[blk * 256 + b * 8 + k],
                candI[blk * 256 + b * 8 + k]);
  }
#pragma unroll
  for (int k = 0; k < 8; ++k) { lv[tid * 8 + k] = tv[k]; li[tid * 8 + k] = ti[k]; }
  __syncthreads();
  if (tid == 0) {
    float fv[8];
    int fi[8];
#pragma unroll
    for (int k = 0; k < 8; ++k) { fv[k] = -3.402823e38f; fi[k] = 0x7fffffff; }
    for (int i = 0; i < 128 * 8; ++i) topk_insert(fv, fi, lv[i], li[i]);
    float m = fv[0], e[8], ssum = 0.f;
#pragma unroll
    for (int k = 0; k < 8; ++k) { e[k] = expf(fv[k] - m); ssum += e[k]; }
    float inv = 1.0f / ssum;
#pragma unroll
    for (int k = 0; k < 8; ++k) { w8[k] = e[k] * inv; i8[k] = fi[k]; }
  }
  __syncthreads();
  float acc = 0.f;
#pragma unroll
  for (int k = 0; k < 8; ++k)
    acc += w8[k] * vault[(long long)i8[k] * DIMS + tid];
  out[b * DIMS + tid] = acc;
}

// ---------------- launch ----------------
extern "C" void kernel_launch(void* const* d_in, const int* in_sizes, int n_in,
                              void* d_out, int out_size, void* d_ws,
                              size_t ws_size, hipStream_t stream) {
  (void)in_sizes; (void)n_in; (void)out_size; (void)ws_size;
  const float* x = (const float*)d_in[0];      // (32,128)
  const float* vault = (const float*)d_in[1];  // (2M,128)
  float* out = (float*)d_out;                  // (32,128)

  float* ws = (float*)d_ws;
  float* bsum = ws;                        // NB_SUM*128      = 131072 f
  float* qext = bsum + NB_SUM * DIMS;      // 80*128          = 10240 f
  float* aq = qext + 80 * DIMS;            // 64 f
  float* candV = aq + 64;                  // NB_SCORE*32*8   = 262144 f
  int* candI = (int*)(candV + NB_SCORE * 32 * 8);  // 262144 i

  mean_kernel<<<NB_SUM, 256, 0, stream>>>(vault, bsum);
  prep_kernel<<<1, 128, 0, stream>>>(x, bsum, qext, aq);
  score_kernel<<<NB_SCORE, 128, 0, stream>>>(vault, qext, aq, candV, candI);
  merge_kernel<<<32, 128, 0, stream>>>(vault, candV, candI, out);
}